// fista_spectral_76038101008534
// MI455X (gfx1250) — compile-verified
//
#include <hip/hip_runtime.h>
#include <cmath>

// ---------------------------------------------------------------------------
// FISTA spectral deconvolution for MI455X (gfx1250, wave32).
// All 2-D FFTs are DFT-matrix GEMMs on the f32 WMMA path
// (V_WMMA_F32_16X16X4_F32), batched over the 32 spectral channels.
// K-panels are staged global->LDS with CDNA5 async copies
// (GLOBAL_LOAD_ASYNC_TO_LDS_B128 / ASYNCcnt), double-buffered.
// ---------------------------------------------------------------------------

typedef float v2f __attribute__((ext_vector_type(2)));
typedef float v8f __attribute__((ext_vector_type(8)));
typedef int   v4i __attribute__((vector_size(16)));

#define NFT   512
#define PLANE (NFT * NFT)        // 262144
#define DD0   256
#define DD1   256
#define NPIXS (DD0 * DD1)        // 65536
#define NCH   32
#define NPSF  3
#define ITERS 10
#define PITER 10

#define WMMA_F32(a, b, c) \
  __builtin_amdgcn_wmma_f32_16x16x4_f32(false, (a), false, (b), (short)0, (c), false, false)

// ---- CDNA5 async global->LDS copy (16B per lane), tracked by ASYNCcnt ----
__device__ __forceinline__ void async_cp16(const float* g, float* l) {
#if __has_builtin(__builtin_amdgcn_global_load_async_to_lds_b128)
  __builtin_amdgcn_global_load_async_to_lds_b128((v4i*)g, (v4i*)l, 0, 0);
#else
  __attribute__((address_space(3))) float* l3 =
      (__attribute__((address_space(3))) float*)l;
  __attribute__((address_space(1))) const float* g1 =
      (__attribute__((address_space(1))) const float*)g;
  asm volatile("global_load_async_to_lds_b128 %0, %1, off" ::"v"(l3), "v"(g1)
               : "memory");
#endif
}

__device__ __forceinline__ void wait_async0() {
#if __has_builtin(__builtin_amdgcn_s_wait_asynccnt)
  __builtin_amdgcn_s_wait_asynccnt(0);
#else
  asm volatile("s_wait_asynccnt 0x0" ::: "memory");
#endif
}

// ===========================================================================
// Complex GEMM, C[b] = A[b] * B[b], 512x512x512, WMMA f32 16x16x4.
// AIM/BIM: A/B have imaginary parts.  WIM: produce imaginary output.
// EPI==1:  C += conj(H) .* (A*B)   (frequency-domain adjoint accumulation)
// Tiling: 64x64 per workgroup; 8 waves, each wave owns one 16-row stripe and
// two 16x16 N-tiles (A fragment reused). BK=32, double-buffered async LDS.
// ===========================================================================
template <bool AIM, bool BIM, bool WIM, int EPI>
__global__ __launch_bounds__(256) void zgemm512(
    const float* __restrict__ Ar_, const float* __restrict__ Ai_, int sA,
    const float* __restrict__ Br_, const float* __restrict__ Bi_, int sB,
    float* __restrict__ Cr_, float* __restrict__ Ci_, int sC,
    const float* __restrict__ Hr_, const float* __restrict__ Hi_) {
  constexpr int PA = AIM ? 2 : 1;         // planes (real[, imag])
  constexpr int PB = BIM ? 2 : 1;
  constexpr int AP = 36;                  // A LDS pitch (16B-aligned rows, conflict-free cols)
  constexpr int BP = 72;                  // B LDS pitch (half-wave K-pairs hit disjoint banks)
  constexpr int ASZ = 64 * AP;
  constexpr int BSZ = 32 * BP;
  __shared__ __align__(16) float lA[2 * PA * ASZ];
  __shared__ __align__(16) float lB[2 * PB * BSZ];

  const int bz = blockIdx.z;
  const float* A_r = Ar_ + (size_t)bz * sA;
  const float* A_i = AIM ? (Ai_ + (size_t)bz * sA) : nullptr;
  const float* B_r = Br_ + (size_t)bz * sB;
  const float* B_i = BIM ? (Bi_ + (size_t)bz * sB) : nullptr;

  const int row0 = blockIdx.x * 64;
  const int col0 = blockIdx.y * 64;
  const int tid  = threadIdx.x;
  const int lane = tid & 31;
  const int w    = tid >> 5;
  const int mbase = (w & 3) << 4;    // 0,16,32,48
  const int nb    = (w >> 2) << 5;   // 0,32  (two 16-wide tiles per wave)
  const int hh    = lane >> 4;       // K-pair half per ISA 16x4 layout
  const int l16   = lane & 15;

  auto loadA = [&](int buf, int k0) {
#pragma unroll
    for (int t = 0; t < 2; ++t) {
      int q = tid + t * 256;                 // 512 16B chunks per plane
      int r = q >> 3, c = (q & 7) << 2;
      size_t g = (size_t)(row0 + r) * NFT + k0 + c;
      async_cp16(A_r + g, &lA[(buf * PA + 0) * ASZ + r * AP + c]);
      if (AIM) async_cp16(A_i + g, &lA[(buf * PA + 1) * ASZ + r * AP + c]);
    }
  };
  auto loadB = [&](int buf, int k0) {
#pragma unroll
    for (int t = 0; t < 2; ++t) {
      int q = tid + t * 256;                 // 512 16B chunks per plane
      int r = q >> 4, c = (q & 15) << 2;
      size_t g = (size_t)(k0 + r) * NFT + col0 + c;
      async_cp16(B_r + g, &lB[(buf * PB + 0) * BSZ + r * BP + c]);
      if (BIM) async_cp16(B_i + g, &lB[(buf * PB + 1) * BSZ + r * BP + c]);
    }
  };

  v8f rr0 = {}, rr1 = {}, ri0 = {}, ri1 = {};
  v8f ir0 = {}, ir1 = {}, ii0 = {}, ii1 = {};

  loadA(0, 0);
  loadB(0, 0);
  wait_async0();
  __syncthreads();

  for (int k0 = 0; k0 < NFT; k0 += 32) {
    const int cur = (k0 >> 5) & 1;
    if (k0 + 32 < NFT) {                 // async prefetch of next K-panel
      loadA(cur ^ 1, k0 + 32);
      loadB(cur ^ 1, k0 + 32);
    }
    const float* Ar0 = &lA[(cur * PA + 0) * ASZ];
    const float* Ai0 = AIM ? &lA[(cur * PA + 1) * ASZ] : nullptr;
    const float* Br0 = &lB[(cur * PB + 0) * BSZ];
    const float* Bi0 = BIM ? &lB[(cur * PB + 1) * BSZ] : nullptr;
#pragma unroll
    for (int kk = 0; kk < 32; kk += 4) {
      const int ko = kk + (hh << 1);
      const int arow = (mbase + l16) * AP;
      const int bidx = ko * BP + nb + l16;
      v2f ar, ai = {}, br0v, br1v, bi0v = {}, bi1v = {};
      ar.x = Ar0[arow + ko];      ar.y = Ar0[arow + ko + 1];
      br0v.x = Br0[bidx];         br0v.y = Br0[bidx + BP];
      br1v.x = Br0[bidx + 16];    br1v.y = Br0[bidx + BP + 16];
      if (AIM) { ai.x = Ai0[arow + ko]; ai.y = Ai0[arow + ko + 1]; }
      if (BIM) {
        bi0v.x = Bi0[bidx];       bi0v.y = Bi0[bidx + BP];
        bi1v.x = Bi0[bidx + 16];  bi1v.y = Bi0[bidx + BP + 16];
      }
      rr0 = WMMA_F32(ar, br0v, rr0);
      rr1 = WMMA_F32(ar, br1v, rr1);
      if (AIM && BIM) { ii0 = WMMA_F32(ai, bi0v, ii0); ii1 = WMMA_F32(ai, bi1v, ii1); }
      if (BIM && WIM) { ri0 = WMMA_F32(ar, bi0v, ri0); ri1 = WMMA_F32(ar, bi1v, ri1); }
      if (AIM && WIM) { ir0 = WMMA_F32(ai, br0v, ir0); ir1 = WMMA_F32(ai, br1v, ir1); }
    }
    wait_async0();
    __syncthreads();
  }

  float* C_r = Cr_ + (size_t)bz * sC;
  float* C_i = WIM ? (Ci_ + (size_t)bz * sC) : nullptr;
#pragma unroll
  for (int i = 0; i < 8; ++i) {
    const int grow = row0 + mbase + (hh << 3) + i;
    const int gc0  = col0 + nb + l16;
    const size_t idx0 = (size_t)grow * NFT + gc0;
    const size_t idx1 = idx0 + 16;
    float cr0 = rr0[i] - ((AIM && BIM) ? ii0[i] : 0.0f);
    float cr1 = rr1[i] - ((AIM && BIM) ? ii1[i] : 0.0f);
    float ci0 = ((BIM && WIM) ? ri0[i] : 0.0f) + ((AIM && WIM) ? ir0[i] : 0.0f);
    float ci1 = ((BIM && WIM) ? ri1[i] : 0.0f) + ((AIM && WIM) ? ir1[i] : 0.0f);
    if (EPI == 0) {
      C_r[idx0] = cr0; C_r[idx1] = cr1;
      if (WIM) { C_i[idx0] = ci0; C_i[idx1] = ci1; }
    } else {  // C += conj(H) * c
      float hr0 = Hr_[idx0], hi0 = Hi_[idx0];
      float hr1 = Hr_[idx1], hi1 = Hi_[idx1];
      C_r[idx0] += hr0 * cr0 + hi0 * ci0;
      C_i[idx0] += hr0 * ci0 - hi0 * cr0;
      C_r[idx1] += hr1 * cr1 + hi1 * ci1;
      C_i[idx1] += hr1 * ci1 - hi1 * cr1;
    }
  }
}

// ===========================================================================
// Elementwise / utility kernels
// ===========================================================================
__global__ void k_init_dft(float* Fr, float* Fi, float* FrI, float* FiI) {
  int idx = blockIdx.x * blockDim.x + threadIdx.x;
  if (idx >= PLANE) return;
  int j = idx >> 9, k = idx & 511;
  int m = (j * k) & 511;
  float t = 6.2831853071795864769f * (float)m / 512.0f;
  float c = cosf(t), s = sinf(t);
  Fr[idx]  = c;                    // exp(-2pi i jk/N)
  Fi[idx]  = -s;
  FrI[idx] = c * (1.0f / 512.0f);  // conj(F)/N
  FiI[idx] = s * (1.0f / 512.0f);
}

__global__ void k_zero(float* p, int n) {
  int i = blockIdx.x * blockDim.x + threadIdx.x;
  if (i < n) p[i] = 0.0f;
}

__global__ void k_build_hpad(const float* __restrict__ h, float* __restrict__ out) {
  int idx = blockIdx.x * blockDim.x + threadIdx.x;
  if (idx >= NPSF * PLANE) return;
  int p = idx / PLANE, r = idx % PLANE;
  int i = r >> 9, j = r & 511;
  int si = (i + 256) & 511, sj = (j + 256) & 511;  // ifftshift of zero-padded PSF
  float v = 0.0f;
  if (si >= 128 && si < 384 && sj >= 128 && sj < 384)
    v = h[p * NPIXS + (si - 128) * DD1 + (sj - 128)];
  out[idx] = v;
}

__global__ void k_masksum(const float* __restrict__ mask, float* __restrict__ msum) {
  int t = blockIdx.x * blockDim.x + threadIdx.x;
  if (t >= NPIXS) return;
  float s = 0.0f;
#pragma unroll
  for (int c = 0; c < NCH; ++c) s += mask[t * NCH + c];
  msum[t] = s;
}

__global__ void k_init_bk(float* bk) {
  int i = blockIdx.x * blockDim.x + threadIdx.x;
  if (i >= PLANE) return;
  float x = sinf((float)i * 12.9898f + 78.233f) * 43758.5453f;
  bk[i] = (x - floorf(x)) - 0.5f;   // deterministic pseudo-random init
}

__global__ void k_ewmul_H(const float* __restrict__ Hr, const float* __restrict__ Hi,
                          const float* __restrict__ Vr, const float* __restrict__ Vi,
                          float* __restrict__ Wr, float* __restrict__ Wi, int n) {
  int idx = blockIdx.x * blockDim.x + threadIdx.x;
  if (idx >= n) return;
  int q = idx & (PLANE - 1);
  float hr = Hr[q], hi = Hi[q], vr = Vr[idx], vi = Vi[idx];
  Wr[idx] = hr * vr - hi * vi;
  Wi[idx] = hr * vi + hi * vr;
}

__global__ void k_err(const float* __restrict__ Ysp, const float* __restrict__ mask,
                      const float* __restrict__ measp, float* __restrict__ errp) {
  int t = blockIdx.x * blockDim.x + threadIdx.x;
  if (t >= NPIXS) return;
  int i = t >> 8, j = t & 255;
  size_t base = (size_t)(i + 128) * NFT + (j + 128);
  float s = 0.0f;
#pragma unroll
  for (int c = 0; c < NCH; ++c) s += mask[t * NCH + c] * Ysp[(size_t)c * PLANE + base];
  errp[t] = s - measp[t];
}

__global__ void k_pow_maskpad(const float* __restrict__ Ysp, const float* __restrict__ msum,
                              float* __restrict__ b1) {
  int idx = blockIdx.x * blockDim.x + threadIdx.x;
  if (idx >= PLANE) return;
  int i = idx >> 9, j = idx & 511;
  float v = 0.0f;
  if (i >= 128 && i < 384 && j >= 128 && j < 384)
    v = Ysp[idx] * msum[(i - 128) * DD1 + (j - 128)];
  b1[idx] = v;
}

__global__ void k_cumsum3(const float* __restrict__ err, float* __restrict__ E) {
  int t = blockIdx.x * blockDim.x + threadIdx.x;
  if (t >= NPIXS) return;
  float e0 = err[t];
  float e1 = e0 + err[NPIXS + t];
  float e2 = e1 + err[2 * NPIXS + t];
  E[t] = e0; E[NPIXS + t] = e1; E[2 * NPIXS + t] = e2;
}

__global__ void k_padmask(const float* __restrict__ Ep, const float* __restrict__ mask,
                          float* __restrict__ Z) {
  int idx = blockIdx.x * blockDim.x + threadIdx.x;
  if (idx >= NCH * PLANE) return;
  int c = idx / PLANE, r = idx % PLANE;
  int i = r >> 9, j = r & 511;
  float v = 0.0f;
  if (i >= 128 && i < 384 && j >= 128 && j < 384) {
    int t = (i - 128) * DD1 + (j - 128);
    v = Ep[t] * mask[t * NCH + c];
  }
  Z[idx] = v;
}

__global__ void k_dot(const float* __restrict__ a, const float* __restrict__ b,
                      float* __restrict__ out, int n) {
  __shared__ float sm[256];
  float s = 0.0f;
  for (int i = blockIdx.x * blockDim.x + threadIdx.x; i < n; i += gridDim.x * blockDim.x)
    s += a[i] * b[i];
  sm[threadIdx.x] = s;
  __syncthreads();
  for (int o = 128; o > 0; o >>= 1) {
    if ((int)threadIdx.x < o) sm[threadIdx.x] += sm[threadIdx.x + o];
    __syncthreads();
  }
  if (threadIdx.x == 0) atomicAdd(out, sm[0]);
}

__global__ void k_scale_rsqrt(float* __restrict__ bk, const float* __restrict__ b1,
                              const float* __restrict__ s) {
  int i = blockIdx.x * blockDim.x + threadIdx.x;
  if (i >= PLANE) return;
  bk[i] = b1[i] * rsqrtf(s[0] + 1e-30f);
}

__global__ void k_compute_L(const float* __restrict__ scal, float* __restrict__ Lout) {
  if (threadIdx.x == 0 && blockIdx.x == 0)
    Lout[0] = 200.0f * scal[0] / scal[1];
}

__global__ void k_prox(const float* __restrict__ g, float* __restrict__ v,
                       float* __restrict__ x, const float* __restrict__ Lp, float tratio) {
  int i = blockIdx.x * blockDim.x + threadIdx.x;
  if (i >= NCH * PLANE) return;
  float inv = 1.0f / (3.0f * Lp[0]);            // grads/P then /L
  float xo = x[i];
  float xn = fmaxf(v[i] - g[i] * inv, 0.0f);
  x[i] = xn;
  v[i] = xn + tratio * (xn - xo);
}

__global__ void k_final_crop(const float* __restrict__ x, float* __restrict__ out) {
  int t = blockIdx.x * blockDim.x + threadIdx.x;
  if (t >= NPIXS * NCH) return;
  int i = t / (DD1 * NCH);
  int rem = t % (DD1 * NCH);
  int j = rem / NCH, c = rem % NCH;
  out[t] = x[(size_t)c * PLANE + (size_t)(i + 128) * NFT + (j + 128)];
}

// ===========================================================================
// Host orchestration
// ===========================================================================
static inline dim3 eb(int n) { return dim3((n + 255) / 256); }

extern "C" void kernel_launch(void* const* d_in, const int* in_sizes, int n_in,
                              void* d_out, int out_size, void* d_ws, size_t ws_size,
                              hipStream_t stream) {
  (void)in_sizes; (void)n_in; (void)out_size; (void)ws_size;
  const float* meas = (const float*)d_in[0];  // [3,256,256]
  const float* h    = (const float*)d_in[1];  // [3,256,256]
  const float* mask = (const float*)d_in[2];  // [256,256,32]
  float* out = (float*)d_out;                 // [256,256,32]

  float* W = (float*)d_ws;
  size_t off = 0;
  auto take = [&](size_t n) { float* p = W + off; off += n; return p; };
  float* Fr  = take(PLANE);          float* Fi  = take(PLANE);
  float* FrI = take(PLANE);          float* FiI = take(PLANE);
  float* Hr  = take(NPSF * PLANE);   float* Hi  = take(NPSF * PLANE);
  float* v   = take(NCH * PLANE);    float* x   = take(NCH * PLANE);
  float* CAr = take(NCH * PLANE);    float* CAi = take(NCH * PLANE);
  float* CBr = take(NCH * PLANE);    float* CBi = take(NCH * PLANE);
  float* CGr = take(NCH * PLANE);    float* CGi = take(NCH * PLANE);
  float* Ysp = take(NCH * PLANE);
  float* bk  = take(PLANE);          float* b1  = take(PLANE);
  float* err = take(NPSF * NPIXS);   float* E   = take(NPSF * NPIXS);
  float* msum = take(NPIXS);
  float* scal = take(8);             // [0],[1]: dots, [2]: L

  const dim3 GB(8, 8, 1);                     // 64x64 tiles, batch via .z
  const dim3 G32(8, 8, NCH);
  const dim3 G3(8, 8, NPSF);
  const dim3 TB(256);

  // ---- setup ----
  k_init_dft<<<eb(PLANE), TB, 0, stream>>>(Fr, Fi, FrI, FiI);
  k_zero<<<eb(NCH * PLANE), TB, 0, stream>>>(v, NCH * PLANE);
  k_zero<<<eb(NCH * PLANE), TB, 0, stream>>>(x, NCH * PLANE);
  k_masksum<<<eb(NPIXS), TB, 0, stream>>>(mask, msum);

  // OTFs: Hs = fft2(ifftshift(pad(h)))   (batch 3, real input)
  k_build_hpad<<<eb(NPSF * PLANE), TB, 0, stream>>>(h, Ysp);
  zgemm512<true, false, true, 0><<<G3, TB, 0, stream>>>(Fr, Fi, 0, Ysp, nullptr, PLANE,
                                                        CAr, CAi, PLANE, nullptr, nullptr);
  zgemm512<true, true, true, 0><<<G3, TB, 0, stream>>>(CAr, CAi, PLANE, Fr, Fi, 0,
                                                       Hr, Hi, PLANE, nullptr, nullptr);

  // ---- power iteration on Hpower (H[0]) ----
  k_init_bk<<<eb(PLANE), TB, 0, stream>>>(bk);
  auto power_step = [&](const float* xin, float* bout) {
    zgemm512<true, false, true, 0><<<GB, TB, 0, stream>>>(Fr, Fi, 0, xin, nullptr, PLANE,
                                                          CAr, CAi, PLANE, nullptr, nullptr);
    zgemm512<true, true, true, 0><<<GB, TB, 0, stream>>>(CAr, CAi, PLANE, Fr, Fi, 0,
                                                         CBr, CBi, PLANE, nullptr, nullptr);
    k_ewmul_H<<<eb(PLANE), TB, 0, stream>>>(Hr, Hi, CBr, CBi, CGr, CGi, PLANE);
    zgemm512<true, true, true, 0><<<GB, TB, 0, stream>>>(FrI, FiI, 0, CGr, CGi, PLANE,
                                                         CAr, CAi, PLANE, nullptr, nullptr);
    zgemm512<true, true, false, 0><<<GB, TB, 0, stream>>>(CAr, CAi, PLANE, FrI, FiI, 0,
                                                          Ysp, nullptr, PLANE, nullptr, nullptr);
    k_pow_maskpad<<<eb(PLANE), TB, 0, stream>>>(Ysp, msum, bout);
  };
  for (int it = 0; it < PITER; ++it) {
    power_step(bk, b1);
    k_zero<<<1, TB, 0, stream>>>(scal, 2);
    k_dot<<<256, TB, 0, stream>>>(b1, b1, scal, PLANE);
    k_scale_rsqrt<<<eb(PLANE), TB, 0, stream>>>(bk, b1, scal);
  }
  power_step(bk, b1);  // Mx in b1
  k_zero<<<1, TB, 0, stream>>>(scal, 2);
  k_dot<<<256, TB, 0, stream>>>(bk, b1, scal, PLANE);      // <bk, Mx>
  k_dot<<<256, TB, 0, stream>>>(bk, bk, scal + 1, PLANE);  // <bk, bk>
  k_compute_L<<<1, 1, 0, stream>>>(scal, scal + 2);

  // ---- FISTA loop ----
  double tk = 1.0;
  for (int it = 0; it < ITERS; ++it) {
    // Vf = fft2(v)  (batch 32, real input)
    zgemm512<true, false, true, 0><<<G32, TB, 0, stream>>>(Fr, Fi, 0, v, nullptr, PLANE,
                                                           CAr, CAi, PLANE, nullptr, nullptr);
    zgemm512<true, true, true, 0><<<G32, TB, 0, stream>>>(CAr, CAi, PLANE, Fr, Fi, 0,
                                                          CBr, CBi, PLANE, nullptr, nullptr);
    // forward model per PSF: err[p] = sum_c mask.*crop(real(ifft2(Hs[p].*Vf))) - meas[p]
    for (int p = 0; p < NPSF; ++p) {
      k_ewmul_H<<<eb(NCH * PLANE), TB, 0, stream>>>(Hr + (size_t)p * PLANE, Hi + (size_t)p * PLANE,
                                                    CBr, CBi, CGr, CGi, NCH * PLANE);
      zgemm512<true, true, true, 0><<<G32, TB, 0, stream>>>(FrI, FiI, 0, CGr, CGi, PLANE,
                                                            CAr, CAi, PLANE, nullptr, nullptr);
      zgemm512<true, true, false, 0><<<G32, TB, 0, stream>>>(CAr, CAi, PLANE, FrI, FiI, 0,
                                                             Ysp, nullptr, PLANE, nullptr, nullptr);
      k_err<<<eb(NPIXS), TB, 0, stream>>>(Ysp, mask, meas + (size_t)p * NPIXS,
                                          err + (size_t)p * NPIXS);
    }
    k_cumsum3<<<eb(NPIXS), TB, 0, stream>>>(err, E);
    // adjoint, accumulated in the frequency domain: Gf = sum_p conj(Hs[p]).*fft2(pad(E[p].*mask))
    k_zero<<<eb(NCH * PLANE), TB, 0, stream>>>(CBr, NCH * PLANE);
    k_zero<<<eb(NCH * PLANE), TB, 0, stream>>>(CBi, NCH * PLANE);
    for (int p = 0; p < NPSF; ++p) {
      k_padmask<<<eb(NCH * PLANE), TB, 0, stream>>>(E + (size_t)p * NPIXS, mask, Ysp);
      zgemm512<true, false, true, 0><<<G32, TB, 0, stream>>>(Fr, Fi, 0, Ysp, nullptr, PLANE,
                                                             CAr, CAi, PLANE, nullptr, nullptr);
      zgemm512<true, true, true, 1><<<G32, TB, 0, stream>>>(CAr, CAi, PLANE, Fr, Fi, 0,
                                                            CBr, CBi, PLANE,
                                                            Hr + (size_t)p * PLANE,
                                                            Hi + (size_t)p * PLANE);
    }
    // grads = real(ifft2(Gf)) / P  (division by P folded into prox)
    zgemm512<true, true, true, 0><<<G32, TB, 0, stream>>>(FrI, FiI, 0, CBr, CBi, PLANE,
                                                          CGr, CGi, PLANE, nullptr, nullptr);
    zgemm512<true, true, false, 0><<<G32, TB, 0, stream>>>(CGr, CGi, PLANE, FrI, FiI, 0,
                                                           Ysp, nullptr, PLANE, nullptr, nullptr);
    // prox + momentum (t-sequence computed on host, deterministic)
    double tup = 1.0 + std::sqrt(1.0 + 4.0 * tk * tk) / 2.0;
    float tratio = (float)((tk - 1.0) / tup);
    k_prox<<<eb(NCH * PLANE), TB, 0, stream>>>(Ysp, v, x, scal + 2, tratio);
    tk = tup;
  }

  k_final_crop<<<eb(NPIXS * NCH), TB, 0, stream>>>(x, out);
}